// GraphSAGEModel_13804024889628
// MI455X (gfx1250) — compile-verified
//
#include <hip/hip_runtime.h>
#include <hip/hip_bf16.h>
#include <math.h>

typedef __attribute__((ext_vector_type(2))) float v2f;
typedef __attribute__((ext_vector_type(8))) float v8f;

#define DIM 128

// ---------------------------------------------------------------- zero scratch
__global__ void gs_zero_f32(float* __restrict__ p, long n) {
    long i = (long)blockIdx.x * blockDim.x + threadIdx.x;
    long stride = (long)gridDim.x * blockDim.x;
    for (; i < n; i += stride) p[i] = 0.0f;
}

// ---------------------------------------------------------------- in-degree
__global__ void gs_degree(const int* __restrict__ ei, int E, float* __restrict__ deg) {
    int e = blockIdx.x * blockDim.x + threadIdx.x;
    if (e < E) atomicAdd(&deg[ei[E + e]], 1.0f);
}

// ---------------------------------------------------------------- layer-1 scatter: agg[dst] += X[src]
// one wave (32 lanes) per edge; each lane moves 4 contiguous channels (float4)
__global__ void gs_scatter_feat(const int* __restrict__ ei, int E,
                                const float4* __restrict__ feat4,
                                float* __restrict__ agg) {
    int t = blockIdx.x * blockDim.x + threadIdx.x;
    int e = t >> 5;
    int lane = t & 31;
    if (e >= E) return;
    int s = ei[e];
    int d = ei[E + e];
    float4 f = feat4[(long)s * 32 + lane];
    float* a = agg + (long)d * DIM + lane * 4;
    atomicAdd(a + 0, f.x);
    atomicAdd(a + 1, f.y);
    atomicAdd(a + 2, f.z);
    atomicAdd(a + 3, f.w);
}

// ---------------------------------------------------------------- layer-1 fused GEMM (fp32 WMMA)
// H = relu(X*Ws + (agg .* inv_deg)*Wn + b1)
// one wave per 16x16 output tile; K-loop over 128 in steps of 4, two
// v_wmma_f32_16x16x4_f32 per step (self path + neighbor path, shared C).
//
// A 16x4 f32 layout (ISA 7.12.2): lanes 0-15 hold (row=lane, K=0..1) in v0..v1,
// lanes 16-31 hold (row=lane-16, K=2..3).  B 4x16 symmetric (col=lane&15,
// K rows split across lane halves).  C/D: VGPR r holds (M=r+8*(lane>=16), N=lane&15).
__global__ __launch_bounds__(32)
void gs_sage1_wmma(const float* __restrict__ X, const float* __restrict__ AGG,
                   const float* __restrict__ DEG,
                   const float* __restrict__ Ws, const float* __restrict__ Wn,
                   const float* __restrict__ B1, float* __restrict__ H, int Nn) {
    int lane = threadIdx.x;
    int m0 = blockIdx.x * 16;
    int n0 = blockIdx.y * 16;
    int half = lane >> 4;
    int l15  = lane & 15;

    int row  = m0 + l15;
    int rowc = row < Nn ? row : (Nn - 1);          // clamp A-reads; stores guarded
    float idg = 1.0f / fmaxf(DEG[rowc], 1.0f);

    int col = n0 + l15;
    float bval = B1[col];
    v8f acc;
#pragma unroll
    for (int r = 0; r < 8; ++r) acc[r] = bval;

    const float* xr = X   + (long)rowc * DIM;
    const float* ar = AGG + (long)rowc * DIM;

#pragma unroll 8
    for (int k = 0; k < DIM; k += 4) {
        int kk = k + 2 * half;
        v2f a_s = *(const v2f*)(xr + kk);
        v2f a_n = *(const v2f*)(ar + kk);
        a_n.x *= idg;
        a_n.y *= idg;
        v2f b_s;
        b_s.x = Ws[(long)kk * DIM + col];
        b_s.y = Ws[(long)(kk + 1) * DIM + col];
        v2f b_n;
        b_n.x = Wn[(long)kk * DIM + col];
        b_n.y = Wn[(long)(kk + 1) * DIM + col];
        acc = __builtin_amdgcn_wmma_f32_16x16x4_f32(false, a_s, false, b_s,
                                                    (short)0, acc, false, false);
        acc = __builtin_amdgcn_wmma_f32_16x16x4_f32(false, a_n, false, b_n,
                                                    (short)0, acc, false, false);
    }

    int rbase = m0 + 8 * half;
#pragma unroll
    for (int r = 0; r < 8; ++r) {
        int orow = rbase + r;
        if (orow < Nn) {
            float v = acc[r];
            H[(long)orow * DIM + col] = v > 0.0f ? v : 0.0f;
        }
    }
}

// ---------------------------------------------------------------- layer-2 projections: sself = h.Ws2, sneigh = h.Wn2
// one wave per node; float4 per lane; shuffle tree reduction (wave32)
__global__ void gs_dot2(const float4* __restrict__ H4,
                        const float4* __restrict__ Ws2,
                        const float4* __restrict__ Wn2,
                        float* __restrict__ sself, float* __restrict__ sneigh, int Nn) {
    int n = blockIdx.x * (blockDim.x >> 5) + (threadIdx.x >> 5);
    int lane = threadIdx.x & 31;
    if (n >= Nn) return;
    float4 h = H4[(long)n * 32 + lane];
    float4 ws = Ws2[lane];
    float4 wn = Wn2[lane];
    float ds = h.x * ws.x + h.y * ws.y + h.z * ws.z + h.w * ws.w;
    float dn = h.x * wn.x + h.y * wn.y + h.z * wn.z + h.w * wn.w;
#pragma unroll
    for (int off = 16; off > 0; off >>= 1) {
        ds += __shfl_xor(ds, off, 32);
        dn += __shfl_xor(dn, off, 32);
    }
    if (lane == 0) {
        sself[n]  = ds;
        sneigh[n] = dn;
    }
}

// ---------------------------------------------------------------- layer-2 scalar scatter: aggs[dst] += sneigh[src]
__global__ void gs_scatter_scalar(const int* __restrict__ ei, int E,
                                  const float* __restrict__ sneigh,
                                  float* __restrict__ aggs) {
    int e = blockIdx.x * blockDim.x + threadIdx.x;
    if (e < E) atomicAdd(&aggs[ei[E + e]], sneigh[ei[e]]);
}

// ---------------------------------------------------------------- epilogue: sigmoid
__global__ void gs_final(const float* __restrict__ sself, const float* __restrict__ aggs,
                         const float* __restrict__ deg, const float* __restrict__ b2,
                         float* __restrict__ out, int Nn) {
    int n = blockIdx.x * blockDim.x + threadIdx.x;
    if (n >= Nn) return;
    float idg = 1.0f / fmaxf(deg[n], 1.0f);
    float x = sself[n] + idg * aggs[n] + b2[0];
    out[n] = 1.0f / (1.0f + expf(-x));
}

// ---------------------------------------------------------------- host
extern "C" void kernel_launch(void* const* d_in, const int* in_sizes, int n_in,
                              void* d_out, int out_size, void* d_ws, size_t ws_size,
                              hipStream_t stream) {
    const float* features = (const float*)d_in[0];
    const int*   ei       = (const int*)d_in[1];     // [2, E] row-major: src then dst
    // d_in[2] = edge_types, unused by the reference
    const float* Ws1 = (const float*)d_in[3];
    const float* Wn1 = (const float*)d_in[4];
    const float* B1  = (const float*)d_in[5];
    const float* Ws2 = (const float*)d_in[6];
    const float* Wn2 = (const float*)d_in[7];
    const float* B2  = (const float*)d_in[8];

    const int Nn = in_sizes[0] / DIM;
    const int E  = in_sizes[1] / 2;

    // workspace layout (floats)
    float* ws = (float*)d_ws;
    float* deg    = ws;                         // [Nn]        (zeroed)
    float* agg1   = deg + Nn;                   // [Nn*DIM]    (zeroed)
    float* aggs   = agg1 + (long)Nn * DIM;      // [Nn]        (zeroed)
    float* h      = aggs + Nn;                  // [Nn*DIM]    (overwritten)
    float* sself  = h + (long)Nn * DIM;         // [Nn]        (overwritten)
    float* sneigh = sself + Nn;                 // [Nn]        (overwritten)

    const long nzero = (long)Nn + (long)Nn * DIM + (long)Nn;  // deg+agg1+aggs contiguous
    gs_zero_f32<<<4096, 256, 0, stream>>>(ws, nzero);

    gs_degree<<<(E + 255) / 256, 256, 0, stream>>>(ei, E, deg);

    {
        long threads = (long)E * 32;
        gs_scatter_feat<<<(unsigned)((threads + 255) / 256), 256, 0, stream>>>(
            ei, E, (const float4*)features, agg1);
    }

    {
        dim3 grid((Nn + 15) / 16, DIM / 16);
        gs_sage1_wmma<<<grid, 32, 0, stream>>>(features, agg1, deg, Ws1, Wn1, B1, h, Nn);
    }

    {
        long threads = (long)Nn * 32;
        gs_dot2<<<(unsigned)((threads + 255) / 256), 256, 0, stream>>>(
            (const float4*)h, (const float4*)Ws2, (const float4*)Wn2, sself, sneigh, Nn);
    }

    gs_scatter_scalar<<<(E + 255) / 256, 256, 0, stream>>>(ei, E, sneigh, aggs);

    gs_final<<<(Nn + 255) / 256, 256, 0, stream>>>(sself, aggs, deg, B2, (float*)d_out, Nn);
}